// LatentMap_5566277616527
// MI455X (gfx1250) — compile-verified
//
#include <hip/hip_runtime.h>

#define IMG   1024
#define NQ    (IMG * IMG)
#define EMB   32
#define KNB   4

typedef float f2 __attribute__((ext_vector_type(2)));
typedef float f4 __attribute__((ext_vector_type(4)));
typedef int   i4 __attribute__((ext_vector_type(4)));

#define HAVE_ASYNC_LDS 0
#define HAVE_WAIT_ASYNC 0
#if defined(__has_builtin)
# if __has_builtin(__builtin_amdgcn_global_load_async_to_lds_b32)
#  undef HAVE_ASYNC_LDS
#  define HAVE_ASYNC_LDS 1
# endif
# if __has_builtin(__builtin_amdgcn_s_wait_asynccnt)
#  undef HAVE_WAIT_ASYNC
#  define HAVE_WAIT_ASYNC 1
# endif
#endif

#if HAVE_ASYNC_LDS
typedef __attribute__((address_space(1))) int gint_t;   // global (device) int
typedef __attribute__((address_space(3))) int lint_t;   // LDS (shared) int

__device__ __forceinline__ gint_t* as_global_int(const float* p) {
    return (gint_t*)(void*)const_cast<float*>(p);
}
__device__ __forceinline__ lint_t* as_lds_int(float* p) {
    return (lint_t*)(void*)p;
}
#endif

__device__ __forceinline__ void wait_async_le1() {
#if HAVE_WAIT_ASYNC
    __builtin_amdgcn_s_wait_asynccnt(1);
#else
    asm volatile("s_wait_asynccnt 0x1" ::: "memory");
#endif
}
__device__ __forceinline__ void wait_async_le0() {
#if HAVE_WAIT_ASYNC
    __builtin_amdgcn_s_wait_asynccnt(0);
#else
    asm volatile("s_wait_asynccnt 0x0" ::: "memory");
#endif
}

// 3-cndmask select of one of 4 wave-uniform values, per-lane index g in [0,3]
__device__ __forceinline__ float sel4f(int g, float a, float b, float c, float d) {
    float lo = (g & 1) ? b : a;
    float hi = (g & 1) ? d : c;
    return (g & 2) ? hi : lo;
}
__device__ __forceinline__ int sel4i(int g, int a, int b, int c, int d) {
    int lo = (g & 1) ? b : a;
    int hi = (g & 1) ? d : c;
    return (g & 2) ? hi : lo;
}

// Process 4 consecutive queries (base..base+3) with one wave.
// p4[j] = position[base+j] (float2).
__device__ __forceinline__ void process4(
    int base, const f2 p4[4], int g, int c, f4 hvec, int lane,
    const f2* __restrict__ pts, const i4* __restrict__ nbm,
    const float* __restrict__ embeddings, float* __restrict__ out)
{
    int ix[4], iy[4];
    i4  nb[4];
#pragma unroll
    for (int j = 0; j < 4; ++j) {
        int x = (int)floorf(p4[j].x);
        int y = (int)floorf(p4[j].y);
        x = x < 0 ? 0 : (x > IMG - 1 ? IMG - 1 : x);
        y = y < 0 ? 0 : (y > IMG - 1 ? IMG - 1 : y);
        ix[j] = __builtin_amdgcn_readfirstlane(x);    // force SGPR -> scalar gathers
        iy[j] = __builtin_amdgcn_readfirstlane(y);
        nb[j] = nbm[(ix[j] << 10) + iy[j]];           // uniform -> s_load_b128
    }

    f2 q[4][KNB];
#pragma unroll
    for (int j = 0; j < 4; ++j) {
#pragma unroll
        for (int k = 0; k < KNB; ++k)
            q[j][k] = pts[nb[j][k]];                  // uniform -> s_load_b64
    }

    const float fx = sel4f(g, (float)ix[0], (float)ix[1], (float)ix[2], (float)ix[3]);
    const float fy = sel4f(g, (float)iy[0], (float)iy[1], (float)iy[2], (float)iy[3]);

    float d[KNB];
    int   nk[KNB];
    float s = 0.0f;
#pragma unroll
    for (int k = 0; k < KNB; ++k) {
        float qx = sel4f(g, q[0][k].x, q[1][k].x, q[2][k].x, q[3][k].x);
        float qy = sel4f(g, q[0][k].y, q[1][k].y, q[2][k].y, q[3][k].y);
        nk[k]    = sel4i(g, nb[0][k], nb[1][k], nb[2][k], nb[3][k]);
        float dx = qx - fx;
        float dy = qy - fy;
        d[k] = sqrtf(fmaf(dx, dx, dy * dy));
        s += d[k];
    }
    const float inv = 1.0f / (s + 1e-8f);

    f4 acc = {0.0f, 0.0f, 0.0f, 0.0f};
#pragma unroll
    for (int k = 0; k < KNB; ++k) {
        const float w = 1.0f - d[k] * inv;
        f4 l = *(const f4*)(embeddings + nk[k] * EMB + c);   // coalesced b128 gather
        acc += l * w;
    }
    f4 res = acc * hvec;

    // one fully coalesced 512B non-temporal store per wave (streamed 128MB output)
    __builtin_nontemporal_store(res, (f4*)(out + (size_t)base * EMB + (size_t)lane * 4));
}

__global__ __launch_bounds__(256) void latent_map_kernel(
    const float* __restrict__ position,      // [NQ, 2]
    const float* __restrict__ positions,     // [NPTS, 2]
    const int*   __restrict__ neighbor_map,  // [IMG, IMG, KNB]
    const float* __restrict__ embeddings,    // [NPTS, EMB]
    const float* __restrict__ harmonics,     // [EMB]
    float*       __restrict__ out)           // [NQ, EMB]
{
    const int lane  = (int)(threadIdx.x & 31u);
    const int wslot = __builtin_amdgcn_readfirstlane((int)(threadIdx.x >> 5));
    const int wave  = (int)blockIdx.x * (int)(blockDim.x >> 5) + wslot;
    const int nw    = (int)(gridDim.x * blockDim.x) >> 5;

    const int g = lane >> 3;          // which of the wave's 4 queries this lane serves
    const int c = (lane & 7) * 4;     // base channel (each lane owns 4 channels)

    const f2* __restrict__ pts = (const f2*)positions;
    const i4* __restrict__ nbm = (const i4*)neighbor_map;

    const f4 hvec = *(const f4*)(harmonics + c);   // cached once per wave

#if HAVE_ASYNC_LDS
    // Per-wave double buffer: 16 queries (32 floats = 128B) per buffer.
    // One GLOBAL_LOAD_ASYNC_TO_LDS_B32 (32 lanes x 4B) fills a buffer one
    // macro-iteration ahead of use; s_wait_asynccnt overlaps copy & compute.
    __shared__ __align__(16) float plds[8][2][32];

    const int mstride = nw * 16;     // queries per macro-iteration across grid
    int m0 = wave * 16;

    if (m0 < NQ) {
        __builtin_amdgcn_global_load_async_to_lds_b32(
            as_global_int(position + (size_t)m0 * 2 + lane),
            as_lds_int(&plds[wslot][0][lane]), 0, 0);
    }

    int buf = 0;
    for (int m = m0; m < NQ; m += mstride, buf ^= 1) {
        const int mn = m + mstride;
        if (mn < NQ) {
            __builtin_amdgcn_global_load_async_to_lds_b32(
                as_global_int(position + (size_t)mn * 2 + lane),
                as_lds_int(&plds[wslot][buf ^ 1][lane]), 0, 0);
            wait_async_le1();        // current buffer complete (in-order), next in flight
        } else {
            wait_async_le0();
        }

#pragma unroll
        for (int j2 = 0; j2 < 4; ++j2) {
            const int base = m + j2 * 4;
            f2 p4[4];
#pragma unroll
            for (int j = 0; j < 4; ++j)
                p4[j] = *(const f2*)&plds[wslot][buf][(j2 * 4 + j) * 2]; // ds_load_b64
            process4(base, p4, g, c, hvec, lane, pts, nbm, embeddings, out);
        }
    }
#else
    const f2* __restrict__ posq = (const f2*)position;
    for (int base = wave * 4; base < NQ; base += nw * 4) {
        if (base + nw * 4 < NQ)
            __builtin_prefetch((const void*)(posq + (base + nw * 4)), 0, 1);
        f2 p4[4];
#pragma unroll
        for (int j = 0; j < 4; ++j)
            p4[j] = posq[base + j];                   // uniform -> s_load_b64
        process4(base, p4, g, c, hvec, lane, pts, nbm, embeddings, out);
    }
#endif
}

extern "C" void kernel_launch(void* const* d_in, const int* in_sizes, int n_in,
                              void* d_out, int out_size, void* d_ws, size_t ws_size,
                              hipStream_t stream) {
    const float* position     = (const float*)d_in[0];
    const float* positions    = (const float*)d_in[1];
    const int*   neighbor_map = (const int*)d_in[2];
    const float* embeddings   = (const float*)d_in[3];
    const float* harmonics    = (const float*)d_in[4];
    float*       out          = (float*)d_out;

    dim3 grid(2048), block(256);   // 16384 waves; 16 queries/wave/macro-iter, 4 macro-iters
    hipLaunchKernelGGL(latent_map_kernel, grid, block, 0, stream,
                       position, positions, neighbor_map, embeddings, harmonics, out);
}